// MultiDimHeadResNet_12446815224304
// MI455X (gfx1250) — compile-verified
//
#include <hip/hip_runtime.h>
#include <math.h>

// ---------------------------------------------------------------------------
// MultiDimHeadResNet decoder for MI455X (gfx1250, wave32, WMMA f32 16x16x4).
//
// d_in order = setup_inputs() dict insertion order, params flattened in
// nested-dict insertion order (see kernel_launch for full map).
// ---------------------------------------------------------------------------

#define BN_EPS 1e-5f

typedef __attribute__((ext_vector_type(2))) float v2f;
typedef __attribute__((ext_vector_type(8))) float v8f;

// Async global->LDS path (gfx1250 GLOBAL_LOAD_ASYNC_TO_LDS_B32, ASYNCcnt).
// Probe-confirmed: builtin exists; param 0 is addrspace(1) int* (HIP prints
// it as '__device__ int *'), param 1 is the LDS dest, then (offset, cpol).
#if __has_builtin(__builtin_amdgcn_global_load_async_to_lds_b32)
#define USE_ASYNC_LDS 1
#else
#define USE_ASYNC_LDS 0
#endif

#if USE_ASYNC_LDS
typedef __attribute__((address_space(1))) int as1_i32;
typedef __attribute__((address_space(3))) int as3_i32;
#define GLOBAL_AS1_I32(p) ((as1_i32*)(void*)(p))
#define LDS_AS3_I32(p)    ((as3_i32*)(void*)(p))
__device__ __forceinline__ void wait_asynccnt0() {
#if __has_builtin(__builtin_amdgcn_s_wait_asynccnt)
  __builtin_amdgcn_s_wait_asynccnt(0);
#else
  asm volatile("s_wait_asynccnt 0x0" ::: "memory");
#endif
}
#endif

// ---------------------------------------------------------------------------
// Masked nearest-neighbor argmin (first-min tie-break like jnp.argmin).
// Masked-out support points get NaN x -> d2 NaN -> never selected.
// ---------------------------------------------------------------------------
__global__ void nn_idx_kernel(const float* __restrict__ qxyz,   // (B,Nq,3)
                              const float* __restrict__ sxyz,   // (B,M,3)
                              const int*   __restrict__ smask,  // (B,M)
                              int*         __restrict__ idx,    // (B,Nq)
                              int Nq, int M) {
  extern __shared__ float s[];  // [3*M]: x | y | z
  const int b = blockIdx.y;
  const float* sp = sxyz + (size_t)b * M * 3;
  const int*   mp = smask + (size_t)b * M;
  for (int j = threadIdx.x; j < M; j += blockDim.x) {
    float x = sp[3 * j + 0];
    float y = sp[3 * j + 1];
    float z = sp[3 * j + 2];
    if (mp[j] <= 0) x = __builtin_nanf("");
    s[j] = x;
    s[M + j] = y;
    s[2 * M + j] = z;
  }
  __syncthreads();

  const int q = blockIdx.x * blockDim.x + threadIdx.x;
  if (q >= Nq) return;
  const float* qp = qxyz + ((size_t)b * Nq + q) * 3;
  const float qx = qp[0], qy = qp[1], qz = qp[2];
  float best = __builtin_inff();
  int bi = 0;
  for (int j = 0; j < M; ++j) {
    const float dx = qx - s[j];
    const float dy = qy - s[M + j];
    const float dz = qz - s[2 * M + j];
    const float d2 = dx * dx + dy * dy + dz * dz;
    if (d2 < best) { best = d2; bi = j; }
  }
  idx[(size_t)b * Nq + q] = bi;
}

// ---------------------------------------------------------------------------
// Fused gather+concat GEMM + BN + ReLU.
//   X[:,n] = concat(prevF[:, idx[n]], lvlF[:, n]);  out = ReLU(bn(W @ X))
//
// Workgroup = 4 waves (128 threads) x one 32-point column tile.
// All threads cooperatively stage a 32(K) x 32(N) chunk of gathered X into
// LDS (via GLOBAL_LOAD_ASYNC_TO_LDS_B32 when available); each wave then runs
// 8 unrolled V_WMMA_F32_16X16X4_F32 steps against two 16-column B fragments
// (ds_load_2addr_b64, conflict-free stride-34 layout), accumulating two
// 16x16 tiles for its own M-tile.
// K-tail (Cin % 32, only the 72-ch head conv) uses the direct global path.
// ---------------------------------------------------------------------------
#define LDS_STRIDE 34  // dwords per staged column (conflict-free, b64-aligned)

__global__ __launch_bounds__(128)
void fused_gather_conv_bn_relu(
    const float* __restrict__ prevF,  // (B,Cprev,Mprev) or nullptr if Cprev==0
    const float* __restrict__ lvlF,   // (B,Cfeat,Nq)
    const int*   __restrict__ idx,    // (B,Nq) or nullptr if Cprev==0
    const float* __restrict__ Wt,     // (Cout,Cin) row-major, Cin=Cprev+Cfeat
    const float* __restrict__ bn_g, const float* __restrict__ bn_b,
    const float* __restrict__ bn_m, const float* __restrict__ bn_v,
    float* __restrict__ out,          // (B,Cout,Nq)
    int Cprev, int Mprev, int Cfeat, int Cout, int Nq) {
  __shared__ float Xs[32 * LDS_STRIDE];

  const int tid = threadIdx.x;
  const int wave = tid >> 5;
  const int lane = tid & 31;
  const int half = lane >> 4;
  const int l = lane & 15;
  const int b = blockIdx.z;
  const int n0 = blockIdx.x << 5;      // 32 points per workgroup
  const int Cin = Cprev + Cfeat;
  const int nMt = (Cout + 15) >> 4;    // number of 16-row M tiles

  // ---- per-thread fill-column setup (column cn of this WG's 32 columns) ----
  const int cn = tid & 31;
  const int krow = tid >> 5;           // 0..3; rows krow, krow+4, ... krow+28
  const float* lvlColFill = lvlF + (size_t)b * Cfeat * Nq + (n0 + cn);
  const float* prevColFill = nullptr;
  if (Cprev > 0) {
    const int g = idx[(size_t)b * Nq + n0 + cn];
    prevColFill = prevF + (size_t)b * Cprev * Mprev + g;
  }
  float* ldsCol = &Xs[cn * LDS_STRIDE];

  // ---- per-wave compute setup ----
  const int mt = blockIdx.y * 4 + wave;          // this wave's M tile
  const int mtc = (mt < nMt) ? mt : (nMt - 1);   // clamp (keeps EXEC uniform)
  const int m0 = mtc << 4;
  int mA = m0 + l;                               // A row fed by this lane
  if (mA >= Cout) mA = Cout - 1;                 // clamp; bad rows not stored
  const float* Arow = Wt + (size_t)mA * Cin;

  v8f acc0 = {};  // columns n0 + 0..15
  v8f acc1 = {};  // columns n0 + 16..31

  const int Cin32 = Cin & ~31;
  for (int k0 = 0; k0 < Cin32; k0 += 32) {
    // -- stage gathered X chunk: 32 K-rows x 32 columns --
#if USE_ASYNC_LDS
#pragma unroll
    for (int i = 0; i < 8; ++i) {
      const int k = krow + 4 * i;
      const int kg = k0 + k;
      const float* src = (kg < Cprev)
                             ? (prevColFill + (size_t)kg * Mprev)
                             : (lvlColFill + (size_t)(kg - Cprev) * Nq);
      __builtin_amdgcn_global_load_async_to_lds_b32(
          GLOBAL_AS1_I32(src), LDS_AS3_I32(ldsCol + k), /*offset=*/0,
          /*cpol=*/0);
    }
    wait_asynccnt0();  // our async LDS writes landed; barrier covers the rest
#else
#pragma unroll
    for (int i = 0; i < 8; ++i) {
      const int k = krow + 4 * i;
      const int kg = k0 + k;
      float v;
      if (kg < Cprev) v = prevColFill[(size_t)kg * Mprev];
      else            v = lvlColFill[(size_t)(kg - Cprev) * Nq];
      ldsCol[k] = v;
    }
#endif
    __builtin_prefetch(Arow + k0 + 32, 0, 1);  // warm next weight chunk
    __syncthreads();

    // -- consume: 8 K-steps x 2 column tiles = 16 WMMA --
#pragma unroll
    for (int j = 0; j < 8; ++j) {
      const int kk = 4 * j + 2 * half;
      v2f a;
      a.x = Arow[k0 + kk];
      a.y = Arow[k0 + kk + 1];
      v2f b0v, b1v;
      b0v.x = Xs[l * LDS_STRIDE + kk];
      b0v.y = Xs[l * LDS_STRIDE + kk + 1];
      b1v.x = Xs[(16 + l) * LDS_STRIDE + kk];
      b1v.y = Xs[(16 + l) * LDS_STRIDE + kk + 1];
      acc0 = __builtin_amdgcn_wmma_f32_16x16x4_f32(false, a, false, b0v,
                                                   (short)0, acc0, false, false);
      acc1 = __builtin_amdgcn_wmma_f32_16x16x4_f32(false, a, false, b1v,
                                                   (short)0, acc1, false, false);
    }
    __syncthreads();
  }

  // -- K tail (Cin % 32; Cprev==0 whenever this runs in this network) --
  if (Cin32 < Cin) {
    const int nA = n0 + l, nB = n0 + 16 + l;
    const float* lvlColA = lvlF + (size_t)b * Cfeat * Nq + nA;
    const float* lvlColB = lvlF + (size_t)b * Cfeat * Nq + nB;
    const float* prevColA = nullptr;
    const float* prevColB = nullptr;
    if (Cprev > 0) {
      prevColA = prevF + (size_t)b * Cprev * Mprev + idx[(size_t)b * Nq + nA];
      prevColB = prevF + (size_t)b * Cprev * Mprev + idx[(size_t)b * Nq + nB];
    }
    for (int k0 = Cin32; k0 < Cin; k0 += 4) {
      const int kk = k0 + 2 * half;
      v2f a;
      a.x = Arow[kk];
      a.y = Arow[kk + 1];
      v2f bA, bB;
      if (k0 < Cprev) {  // uniform: Cprev multiple of 4
        bA.x = prevColA[(size_t)kk * Mprev];
        bA.y = prevColA[(size_t)(kk + 1) * Mprev];
        bB.x = prevColB[(size_t)kk * Mprev];
        bB.y = prevColB[(size_t)(kk + 1) * Mprev];
      } else {
        const int kf = kk - Cprev;
        bA.x = lvlColA[(size_t)kf * Nq];
        bA.y = lvlColA[(size_t)(kf + 1) * Nq];
        bB.x = lvlColB[(size_t)kf * Nq];
        bB.y = lvlColB[(size_t)(kf + 1) * Nq];
      }
      acc0 = __builtin_amdgcn_wmma_f32_16x16x4_f32(false, a, false, bA,
                                                   (short)0, acc0, false, false);
      acc1 = __builtin_amdgcn_wmma_f32_16x16x4_f32(false, a, false, bB,
                                                   (short)0, acc1, false, false);
    }
  }

  // ---- folded BatchNorm + ReLU epilogue, both column tiles ----
  if (mt < nMt) {
#pragma unroll
    for (int r = 0; r < 8; ++r) {
      const int m = m0 + r + 8 * half;
      if (m < Cout) {
        const float sc = bn_g[m] / sqrtf(bn_v[m] + BN_EPS);
        const float sh = bn_b[m] - bn_m[m] * sc;
        float* orow = out + ((size_t)b * Cout + m) * Nq;
        float y0 = fmaxf(acc0[r] * sc + sh, 0.0f);
        float y1 = fmaxf(acc1[r] * sc + sh, 0.0f);
        orow[n0 + l] = y0;
        orow[n0 + 16 + l] = y1;
      }
    }
  }
}

// ---------------------------------------------------------------------------
// Final head: out[b,o,n] = sum_c Wh2[o,c] * h[b,c,n] + bh2[o]   (o < 3)
// ---------------------------------------------------------------------------
__global__ void head_out_kernel(const float* __restrict__ h,    // (B,72,Nq)
                                const float* __restrict__ W2,   // (3,72)
                                const float* __restrict__ b2,   // (3,)
                                float* __restrict__ out,        // (B,3,Nq)
                                int Nq) {
  const int n = blockIdx.x * blockDim.x + threadIdx.x;
  const int b = blockIdx.y;
  if (n >= Nq) return;
  const float* hc = h + (size_t)b * 72 * Nq + n;
  float a0 = b2[0], a1 = b2[1], a2 = b2[2];
  for (int c = 0; c < 72; ++c) {
    const float v = hc[(size_t)c * Nq];
    a0 = fmaf(W2[0 * 72 + c], v, a0);
    a1 = fmaf(W2[1 * 72 + c], v, a1);
    a2 = fmaf(W2[2 * 72 + c], v, a2);
  }
  out[((size_t)b * 3 + 0) * Nq + n] = a0;
  out[((size_t)b * 3 + 1) * Nq + n] = a1;
  out[((size_t)b * 3 + 2) * Nq + n] = a2;
}

// ---------------------------------------------------------------------------
extern "C" void kernel_launch(void* const* d_in, const int* in_sizes, int n_in,
                              void* d_out, int out_size, void* d_ws, size_t ws_size,
                              hipStream_t stream) {
  (void)in_sizes; (void)n_in; (void)out_size; (void)ws_size;
  const int B = 8;
  const int N1 = 8192, N2 = 2048, N3 = 512, N4 = 128, N5 = 32;
  const int C1 = 144, C2 = 288, C3 = 576, C4 = 1152, C5 = 2304;

  // d_in map: per level l=1..5: [3(l-1)+0]=xyz, [+1]=mask, [+2]=features;
  // then params: W0,bn0(g,b,m,v), W1,bn1, W2,bn2, W3,bn3, Wh1,bnh, Wh2,bh2.
  const float* xyz1 = (const float*)d_in[0];
  const int*   msk1 = (const int*)  d_in[1];  (void)msk1;
  const float* ft1  = (const float*)d_in[2];
  const float* xyz2 = (const float*)d_in[3];
  const int*   msk2 = (const int*)  d_in[4];
  const float* ft2  = (const float*)d_in[5];
  const float* xyz3 = (const float*)d_in[6];
  const int*   msk3 = (const int*)  d_in[7];
  const float* ft3  = (const float*)d_in[8];
  const float* xyz4 = (const float*)d_in[9];
  const int*   msk4 = (const int*)  d_in[10];
  const float* ft4  = (const float*)d_in[11];
  const float* xyz5 = (const float*)d_in[12];
  const int*   msk5 = (const int*)  d_in[13];
  const float* ft5  = (const float*)d_in[14];

  const float* W0 = (const float*)d_in[15];
  const float* g0 = (const float*)d_in[16]; const float* b0 = (const float*)d_in[17];
  const float* m0 = (const float*)d_in[18]; const float* v0 = (const float*)d_in[19];
  const float* W1 = (const float*)d_in[20];
  const float* g1 = (const float*)d_in[21]; const float* b1 = (const float*)d_in[22];
  const float* m1 = (const float*)d_in[23]; const float* v1 = (const float*)d_in[24];
  const float* W2 = (const float*)d_in[25];
  const float* g2 = (const float*)d_in[26]; const float* b2 = (const float*)d_in[27];
  const float* m2 = (const float*)d_in[28]; const float* v2 = (const float*)d_in[29];
  const float* W3 = (const float*)d_in[30];
  const float* g3 = (const float*)d_in[31]; const float* b3 = (const float*)d_in[32];
  const float* m3 = (const float*)d_in[33]; const float* v3 = (const float*)d_in[34];
  const float* Wh1 = (const float*)d_in[35];
  const float* gh = (const float*)d_in[36]; const float* bh = (const float*)d_in[37];
  const float* mh = (const float*)d_in[38]; const float* vh = (const float*)d_in[39];
  const float* Wh2 = (const float*)d_in[40];
  const float* bh2 = (const float*)d_in[41];

  float* out = (float*)d_out;

  // ---- workspace layout ----
  char* ws = (char*)d_ws;
  size_t off = 0;
  int* idx4 = (int*)(ws + off); off += (size_t)B * N4 * sizeof(int);
  int* idx3 = (int*)(ws + off); off += (size_t)B * N3 * sizeof(int);
  int* idx2 = (int*)(ws + off); off += (size_t)B * N2 * sizeof(int);
  int* idx1 = (int*)(ws + off); off += (size_t)B * N1 * sizeof(int);
  float* f4 = (float*)(ws + off); off += (size_t)B * 576 * N4 * sizeof(float);
  float* f3 = (float*)(ws + off); off += (size_t)B * 288 * N3 * sizeof(float);
  float* f2 = (float*)(ws + off); off += (size_t)B * 144 * N2 * sizeof(float);
  float* f1 = (float*)(ws + off); off += (size_t)B * 72 * N1 * sizeof(float);
  float* hh = (float*)(ws + off); off += (size_t)B * 72 * N1 * sizeof(float);
  // total ~54.6 MB

  // ---- stage 5 -> 4 ----
  nn_idx_kernel<<<dim3((N4 + 255) / 256, B), 256, 3 * N5 * sizeof(float), stream>>>(
      xyz4, xyz5, msk5, idx4, N4, N5);
  fused_gather_conv_bn_relu<<<dim3(N4 / 32, 9, B), 128, 0, stream>>>(
      ft5, ft4, idx4, W0, g0, b0, m0, v0, f4,
      /*Cprev=*/C5, /*Mprev=*/N5, /*Cfeat=*/C4, /*Cout=*/576, /*Nq=*/N4);

  // ---- stage 4 -> 3 ----
  nn_idx_kernel<<<dim3((N3 + 255) / 256, B), 256, 3 * N4 * sizeof(float), stream>>>(
      xyz3, xyz4, msk4, idx3, N3, N4);
  fused_gather_conv_bn_relu<<<dim3(N3 / 32, 5, B), 128, 0, stream>>>(
      f4, ft3, idx3, W1, g1, b1, m1, v1, f3,
      /*Cprev=*/576, /*Mprev=*/N4, /*Cfeat=*/C3, /*Cout=*/288, /*Nq=*/N3);

  // ---- stage 3 -> 2 ----
  nn_idx_kernel<<<dim3((N2 + 255) / 256, B), 256, 3 * N3 * sizeof(float), stream>>>(
      xyz2, xyz3, msk3, idx2, N2, N3);
  fused_gather_conv_bn_relu<<<dim3(N2 / 32, 3, B), 128, 0, stream>>>(
      f3, ft2, idx2, W2, g2, b2, m2, v2, f2,
      /*Cprev=*/288, /*Mprev=*/N3, /*Cfeat=*/C2, /*Cout=*/144, /*Nq=*/N2);

  // ---- stage 2 -> 1 ----
  nn_idx_kernel<<<dim3((N1 + 255) / 256, B), 256, 3 * N2 * sizeof(float), stream>>>(
      xyz1, xyz2, msk2, idx1, N1, N2);
  fused_gather_conv_bn_relu<<<dim3(N1 / 32, 2, B), 128, 0, stream>>>(
      f2, ft1, idx1, W3, g3, b3, m3, v3, f1,
      /*Cprev=*/144, /*Mprev=*/N2, /*Cfeat=*/C1, /*Cout=*/72, /*Nq=*/N1);

  // ---- head conv (no gather/concat; Cin=72 exercises the K-tail path) ----
  fused_gather_conv_bn_relu<<<dim3(N1 / 32, 2, B), 128, 0, stream>>>(
      nullptr, f1, nullptr, Wh1, gh, bh, mh, vh, hh,
      /*Cprev=*/0, /*Mprev=*/1, /*Cfeat=*/72, /*Cout=*/72, /*Nq=*/N1);

  // ---- final 3x72 projection + bias ----
  head_out_kernel<<<dim3(N1 / 256, B), 256, 0, stream>>>(hh, Wh2, bh2, out, N1);
}